// ShapeletNet_63015760167422
// MI455X (gfx1250) — compile-verified
//
#include <hip/hip_runtime.h>
#include <hip/hip_bf16.h>
#include <math.h>

typedef float v2f __attribute__((ext_vector_type(2)));
typedef float v8f __attribute__((ext_vector_type(8)));

#define B_      256
#define L_      2048
#define V_      8
#define S_      5
#define BAG_    409
#define NB_     9
#define SHIFT_  204
#define KPAD_   412          // BAG padded up to multiple of 4
#define FEAT_   80
#define NCLASS_ 10

// ---- workspace layout (float offsets) ----
#define WS_BPAD  0                              // [V][KPAD][16] padded shapelet B-matrices
#define WS_S2    (WS_BPAD + V_*KPAD_*16)        // [V][16]  sum of squares per (v, s)
#define WS_DBUF  (WS_S2 + V_*16)                // [B][S][V][NB] per-window distances
#define WS_FEAT  (WS_DBUF + B_*S_*V_*NB_)       // [B][80]  (0..39 = min, 40..79 = mean)
#define WS_WP    (WS_FEAT + B_*FEAT_)           // [80][16] BN-folded FC weights (K-major)
#define WS_BIAS  (WS_WP + FEAT_*16)             // [16]     BN-folded FC bias

// ------------------------------------------------------------------
// k0: pack shapelets into WMMA-B layout ([V][kk][n], zero padded) and
//     precompute S2[v][s] = sum_k shp[s,v,k]^2
// ------------------------------------------------------------------
__global__ void k0_prep(const float* __restrict__ shp, float* __restrict__ ws) {
  const int stride = gridDim.x * blockDim.x;
  const int tid0 = blockIdx.x * blockDim.x + threadIdx.x;

  for (int idx = tid0; idx < V_ * KPAD_ * 16; idx += stride) {
    const int v  = idx / (KPAD_ * 16);
    const int r  = idx - v * (KPAD_ * 16);
    const int kk = r >> 4;
    const int n  = r & 15;
    float val = 0.0f;
    if (n < S_ && kk < BAG_) val = shp[n * (V_ * BAG_) + v * BAG_ + kk];
    ws[WS_BPAD + idx] = val;
  }
  for (int idx = tid0; idx < V_ * 16; idx += stride) {
    const int v = idx >> 4;
    const int n = idx & 15;
    float acc = 0.0f;
    if (n < S_) {
      const float* p = shp + n * (V_ * BAG_) + v * BAG_;
      for (int k = 0; k < BAG_; ++k) { float t = p[k]; acc += t * t; }
    }
    ws[WS_S2 + idx] = acc;
  }
}

// ------------------------------------------------------------------
// k1: one wave per (b-tile of 16, window nb, variable v).
//     C_cross += A * B_shapelet   (fp32 WMMA 16x16x4)
//     C_x2    += (A.*A) * Ones    (window energy in every column)
//     dist = sqrt(X2 - 2*cross + S2)  -> DBUF[b][s][v][nb]
// ------------------------------------------------------------------
__global__ void __launch_bounds__(32)
k1_dist(const float* __restrict__ x, float* __restrict__ ws) {
  const int bt   = blockIdx.x;   // 0..15  (batch tile)
  const int nb   = blockIdx.y;   // 0..8   (window)
  const int v    = blockIdx.z;   // 0..7   (variable)
  const int lane = threadIdx.x;  // 0..31
  const int s    = lane & 15;    // column / row-in-tile id
  const int hi   = lane >> 4;    // 0: K={k,k+1}, 1: K={k+2,k+3}

  const int b0    = bt * 16;
  const int start = nb * SHIFT_;

  // A: row M=s of x-window (x buffer reinterpreted as [B][V][L])
  const float* arow = x + (b0 + s) * (V_ * L_) + v * L_ + start + 2 * hi;
  // B: shapelet matrix, column N=s
  const float* bcol = ws + WS_BPAD + v * (KPAD_ * 16) + (2 * hi) * 16 + s;

  v8f c_cross = {};
  v8f c_x2 = {};
  const v2f ones = {1.0f, 1.0f};

  // 102 full K=4 steps covering kk = 0..407
#pragma unroll 2
  for (int k = 0; k < 408; k += 4) {
    v2f a = *(const v2f*)(arow + k);          // 8B-aligned pair load
    v2f b;
    b.x = bcol[k * 16];
    b.y = bcol[k * 16 + 16];
    v2f a2 = a * a;
    c_cross = __builtin_amdgcn_wmma_f32_16x16x4_f32(
        false, a, false, b, (short)0, c_cross, false, false);
    c_x2 = __builtin_amdgcn_wmma_f32_16x16x4_f32(
        false, a2, false, ones, (short)0, c_x2, false, false);
  }
  // tail step: only kk = 408 is valid; mask via B operands (Bpad rows >=409 are 0)
  {
    const int k = 408;
    v2f a = *(const v2f*)(arow + k);          // in-bounds (start+411 <= 2043)
    v2f b;
    b.x = bcol[k * 16];
    b.y = bcol[k * 16 + 16];
    v2f a2 = a * a;
    v2f bones;
    bones.x = hi ? 0.0f : 1.0f;               // kk=408 valid, 410 not
    bones.y = 0.0f;                           // kk=409/411 not
    c_cross = __builtin_amdgcn_wmma_f32_16x16x4_f32(
        false, a, false, b, (short)0, c_cross, false, false);
    c_x2 = __builtin_amdgcn_wmma_f32_16x16x4_f32(
        false, a2, false, bones, (short)0, c_x2, false, false);
  }

  // epilogue: C element j in this lane is (M = j + 8*hi, N = s)
  const float s2 = ws[WS_S2 + v * 16 + s];
  if (s < S_) {
#pragma unroll
    for (int j = 0; j < 8; ++j) {
      const int m = j + 8 * hi;
      const int b = b0 + m;
      float d2 = c_x2[j] - 2.0f * c_cross[j] + s2;
      float d = sqrtf(fmaxf(d2, 0.0f));
      ws[WS_DBUF + ((b * S_ + s) * V_ + v) * NB_ + nb] = d;
    }
  }
}

// ------------------------------------------------------------------
// k2a: reduce the 9 windows -> feat[b][f]  (min | mean), deterministic
// ------------------------------------------------------------------
__global__ void k2a_reduce(float* __restrict__ ws) {
  const int idx = blockIdx.x * blockDim.x + threadIdx.x;
  if (idx >= B_ * S_ * V_) return;
  const int b  = idx / (S_ * V_);
  const int sv = idx - b * (S_ * V_);
  const float* d = ws + WS_DBUF + idx * NB_;
  float mn = d[0], sum = d[0];
#pragma unroll
  for (int nb = 1; nb < NB_; ++nb) {
    float t = d[nb];
    mn = fminf(mn, t);
    sum += t;
  }
  ws[WS_FEAT + b * FEAT_ + sv] = mn;
  ws[WS_FEAT + b * FEAT_ + 40 + sv] = sum * (1.0f / NB_);
}

// ------------------------------------------------------------------
// k2b: batch stats + fold BatchNorm into FC:  W'[f][n] = w[n,f]*scale[f]
//      bias'[n] = fc_b[n] + sum_f shift[f]*w[n,f]
// ------------------------------------------------------------------
__global__ void __launch_bounds__(128)
k2b_stats(const float* __restrict__ gamma, const float* __restrict__ beta,
          const float* __restrict__ fcw, const float* __restrict__ fcb,
          float* __restrict__ ws) {
  __shared__ float Sc[FEAT_];
  __shared__ float Sh[FEAT_];
  const int t = threadIdx.x;
  if (t < FEAT_) {
    const float* feat = ws + WS_FEAT;
    float sum = 0.0f, ss = 0.0f;
    for (int b = 0; b < B_; ++b) {
      float val = feat[b * FEAT_ + t];
      sum += val;
      ss += val * val;
    }
    const float mu = sum * (1.0f / B_);
    const float var = ss * (1.0f / B_) - mu * mu;  // biased variance
    const float rstd = rsqrtf(var + 1e-5f);
    const float g = gamma[t] * rstd;
    Sc[t] = g;
    Sh[t] = beta[t] - mu * g;
  }
  __syncthreads();
  for (int idx = t; idx < FEAT_ * 16; idx += blockDim.x) {
    const int f = idx >> 4;
    const int n = idx & 15;
    ws[WS_WP + idx] = (n < NCLASS_) ? fcw[n * FEAT_ + f] * Sc[f] : 0.0f;
  }
  if (t < 16) {
    float bp = 0.0f;
    if (t < NCLASS_) {
      bp = fcb[t];
      for (int f = 0; f < FEAT_; ++f) bp += Sh[f] * fcw[t * FEAT_ + f];
    }
    ws[WS_BIAS + t] = bp;
  }
}

// ------------------------------------------------------------------
// k3: FC as fp32 WMMA GEMM: out[256,10] = feat[256,80] @ W'[80,16] + b'
// ------------------------------------------------------------------
__global__ void __launch_bounds__(32)
k3_fc(const float* __restrict__ ws, float* __restrict__ out) {
  const int lane = threadIdx.x;
  const int s = lane & 15;
  const int hi = lane >> 4;
  const int b0 = blockIdx.x * 16;

  const float* arow = ws + WS_FEAT + (b0 + s) * FEAT_ + 2 * hi;
  const float* bcol = ws + WS_WP + (2 * hi) * 16 + s;

  v8f c = {};
#pragma unroll
  for (int k = 0; k < FEAT_; k += 4) {
    v2f a = *(const v2f*)(arow + k);
    v2f b;
    b.x = bcol[k * 16];
    b.y = bcol[k * 16 + 16];
    c = __builtin_amdgcn_wmma_f32_16x16x4_f32(
        false, a, false, b, (short)0, c, false, false);
  }

  const float bias = ws[WS_BIAS + s];
  if (s < NCLASS_) {
#pragma unroll
    for (int j = 0; j < 8; ++j) {
      const int m = j + 8 * hi;
      out[(b0 + m) * NCLASS_ + s] = c[j] + bias;
    }
  }
}

// ------------------------------------------------------------------
extern "C" void kernel_launch(void* const* d_in, const int* in_sizes, int n_in,
                              void* d_out, int out_size, void* d_ws, size_t ws_size,
                              hipStream_t stream) {
  const float* x     = (const float*)d_in[0];
  const float* shp   = (const float*)d_in[1];
  const float* gamma = (const float*)d_in[2];
  const float* beta  = (const float*)d_in[3];
  const float* fcw   = (const float*)d_in[4];
  const float* fcb   = (const float*)d_in[5];
  float* out = (float*)d_out;
  float* ws  = (float*)d_ws;

  k0_prep<<<64, 256, 0, stream>>>(shp, ws);
  k1_dist<<<dim3(16, 9, 8), 32, 0, stream>>>(x, ws);
  k2a_reduce<<<(B_ * S_ * V_ + 255) / 256, 256, 0, stream>>>(ws);
  k2b_stats<<<1, 128, 0, stream>>>(gamma, beta, fcw, fcb, ws);
  k3_fc<<<16, 32, 0, stream>>>(ws, out);
}